// PoseGraph_30399778521135
// MI455X (gfx1250) — compile-verified
//
#include <hip/hip_runtime.h>
#include <math.h>
#include <stdint.h>

#define BLOCK 256
#define EPSF 1e-6f

struct V3 { float x, y, z; };
struct Q4 { float x, y, z, w; };
struct SE3 { V3 t; Q4 q; };

__device__ __forceinline__ V3 cross3(V3 a, V3 b) {
    return { a.y * b.z - a.z * b.y,
             a.z * b.x - a.x * b.z,
             a.x * b.y - a.y * b.x };
}

// v + 2*cross(qv, cross(qv,v) + w*v)
__device__ __forceinline__ V3 qrot(Q4 q, V3 v) {
    V3 qv { q.x, q.y, q.z };
    V3 c = cross3(qv, v);
    V3 t { c.x + q.w * v.x, c.y + q.w * v.y, c.z + q.w * v.z };
    V3 c2 = cross3(qv, t);
    return { v.x + 2.0f * c2.x, v.y + 2.0f * c2.y, v.z + 2.0f * c2.z };
}

__device__ __forceinline__ Q4 qmul(Q4 a, Q4 b) {
    return { a.w * b.x + a.x * b.w + a.y * b.z - a.z * b.y,
             a.w * b.y - a.x * b.z + a.y * b.w + a.z * b.x,
             a.w * b.z + a.x * b.y - a.y * b.x + a.z * b.w,
             a.w * b.w - a.x * b.x - a.y * b.y - a.z * b.z };
}

__device__ __forceinline__ SE3 se3_inv(SE3 T) {
    Q4 qi { -T.q.x, -T.q.y, -T.q.z, T.q.w };
    V3 r = qrot(qi, T.t);
    return { { -r.x, -r.y, -r.z }, qi };
}

__device__ __forceinline__ SE3 se3_mul(SE3 A, SE3 B) {
    V3 rt = qrot(A.q, B.t);
    return { { A.t.x + rt.x, A.t.y + rt.y, A.t.z + rt.z }, qmul(A.q, B.q) };
}

__device__ __forceinline__ V3 so3_log(Q4 q) {
    float n2 = q.x * q.x + q.y * q.y + q.z * q.z;
    float n = sqrtf(n2);
    float scale;
    if (n < EPSF) {
        float w3 = q.w * q.w * q.w;
        scale = 2.0f / q.w - 2.0f * n2 / (3.0f * w3);
    } else {
        scale = 2.0f * atan2f(n, q.w) / n;
    }
    return { scale * q.x, scale * q.y, scale * q.z };
}

__device__ __forceinline__ void se3_log(SE3 T, float xi[6]) {
    V3 phi = so3_log(T.q);
    float th2 = phi.x * phi.x + phi.y * phi.y + phi.z * phi.z;
    float th = sqrtf(th2);
    float coef;
    if (th < EPSF) {
        coef = 1.0f / 12.0f;
    } else {
        coef = 1.0f / th2 - (1.0f + cosf(th)) / (2.0f * th * sinf(th));
    }
    V3 pxt  = cross3(phi, T.t);
    V3 ppxt = cross3(phi, pxt);
    xi[0] = T.t.x - 0.5f * pxt.x + coef * ppxt.x;
    xi[1] = T.t.y - 0.5f * pxt.y + coef * ppxt.y;
    xi[2] = T.t.z - 0.5f * pxt.z + coef * ppxt.z;
    xi[3] = phi.x;
    xi[4] = phi.y;
    xi[5] = phi.z;
}

__global__ __launch_bounds__(BLOCK) void posegraph_kernel(
    const int*   __restrict__ edges,   // (E,2) int32
    const float* __restrict__ poses,   // (E,7)
    const float* __restrict__ infos,   // (E,6,6)
    const float* __restrict__ nodes,   // (N,7)
    float*       __restrict__ out,     // (E,6)
    int nEdges)
{
    // 256 edges * 36 f32 = 36 KB staged per workgroup (LDS budget: 320 KB/WGP)
    __shared__ float smA[BLOCK * 36];

    const int tid = threadIdx.x;
    const int e0  = (int)blockIdx.x * BLOCK;
    int nBlk = nEdges - e0;
    if (nBlk > BLOCK) nBlk = BLOCK;

    // ---- async DMA: stage this block's info matrices global -> LDS --------
    // 16B chunks, lane-contiguous => each wave issues 512B fully-coalesced
    // transfers tracked on ASYNCcnt, overlapped with the SE3 math below.
    // GVS addressing: uniform tile base in SGPR pair, 32-bit lane offset in
    // a VGPR (no 64-bit per-lane address math). chunk c = tid + k*256 lives
    // at byte offset c*16 in BOTH the LDS tile and the global tile, and
    // INST_OFFSET is added to both sides (ISA 08 §4.4) -> full blocks are
    // 9 immediate-offset instructions issued as one clause.
    uint32_t lbase = (uint32_t)(uintptr_t)smA;           // low 32b = LDS addr
    const char* gsrc = (const char*)infos + (size_t)e0 * 144u;
    if (nBlk == BLOCK) {
        uint32_t voff = (uint32_t)tid * 16u;             // lane offset
        uint32_t loff = lbase + voff;                    // LDS dest addr
        asm volatile(
            "s_clause 0x8\n\t"
            "global_load_async_to_lds_b128 %0, %1, %2\n\t"
            "global_load_async_to_lds_b128 %0, %1, %2 offset:4096\n\t"
            "global_load_async_to_lds_b128 %0, %1, %2 offset:8192\n\t"
            "global_load_async_to_lds_b128 %0, %1, %2 offset:12288\n\t"
            "global_load_async_to_lds_b128 %0, %1, %2 offset:16384\n\t"
            "global_load_async_to_lds_b128 %0, %1, %2 offset:20480\n\t"
            "global_load_async_to_lds_b128 %0, %1, %2 offset:24576\n\t"
            "global_load_async_to_lds_b128 %0, %1, %2 offset:28672\n\t"
            "global_load_async_to_lds_b128 %0, %1, %2 offset:32768"
            :: "v"(loff), "v"(voff), "s"(gsrc) : "memory");
    } else {
        const int nChunks = nBlk * 9;                    // 144B/edge = 9 chunks
        for (int c = tid; c < nChunks; c += BLOCK) {
            uint32_t voff = (uint32_t)c * 16u;
            uint32_t loff = lbase + voff;
            asm volatile("global_load_async_to_lds_b128 %0, %1, %2"
                         :: "v"(loff), "v"(voff), "s"(gsrc) : "memory");
        }
    }

    // ---- per-edge SE3 chain + log (register math, hides DMA latency) ------
    float xi[6];
    const bool active = (tid < nBlk);
    if (active) {
        const int e  = e0 + tid;
        const int i1 = __builtin_nontemporal_load(&edges[2 * e + 0]);
        const int i2 = __builtin_nontemporal_load(&edges[2 * e + 1]);

        const float* p = poses + (size_t)e * 7;
        SE3 Z  { { __builtin_nontemporal_load(p + 0),
                   __builtin_nontemporal_load(p + 1),
                   __builtin_nontemporal_load(p + 2) },
                 { __builtin_nontemporal_load(p + 3),
                   __builtin_nontemporal_load(p + 4),
                   __builtin_nontemporal_load(p + 5),
                   __builtin_nontemporal_load(p + 6) } };

        // node gathers: 5.6 MB table, random index -> serviced by 192 MB L2
        const float* a = nodes + (size_t)i1 * 7;
        SE3 N1 { { a[0], a[1], a[2] }, { a[3], a[4], a[5], a[6] } };
        const float* b = nodes + (size_t)i2 * 7;
        SE3 N2 { { b[0], b[1], b[2] }, { b[3], b[4], b[5], b[6] } };

        SE3 T = se3_mul(se3_mul(se3_inv(Z), se3_inv(N1)), N2);
        se3_log(T, xi);
    }

    // ---- wait for this wave's async transfers, then workgroup barrier -----
    asm volatile("s_wait_asynccnt 0x0" ::: "memory");
    __syncthreads();

    // ---- 6x6 matvec from LDS, non-temporal streaming store ----------------
    if (active) {
        const int e = e0 + tid;
        const float* M = &smA[tid * 36];
        float* o = out + (size_t)e * 6;
        #pragma unroll
        for (int i = 0; i < 6; ++i) {
            float acc = 0.0f;
            #pragma unroll
            for (int j = 0; j < 6; ++j)
                acc += M[i * 6 + j] * xi[j];
            __builtin_nontemporal_store(acc, &o[i]);
        }
    }
}

extern "C" void kernel_launch(void* const* d_in, const int* in_sizes, int n_in,
                              void* d_out, int out_size, void* d_ws, size_t ws_size,
                              hipStream_t stream) {
    const int*   edges = (const int*)d_in[0];
    const float* poses = (const float*)d_in[1];
    const float* infos = (const float*)d_in[2];
    const float* nodes = (const float*)d_in[3];
    float* out = (float*)d_out;

    const int nEdges = in_sizes[0] / 2;
    const int grid = (nEdges + BLOCK - 1) / BLOCK;
    hipLaunchKernelGGL(posegraph_kernel, dim3(grid), dim3(BLOCK), 0, stream,
                       edges, poses, infos, nodes, out, nEdges);
}